// AlphaChebyshevProjection_84000970375198
// MI455X (gfx1250) — compile-verified
//
#include <hip/hip_runtime.h>

// ---- problem constants (match reference) ----
#define Mm    512      // constraints per problem
#define Nn    128      // variables
#define NZ    129      // z = (x, r)
#define NTI   9        // 9 tiles of 16 -> 144 padded cols
#define HS    131      // H row stride (floats), gcd(131,64)=1
#define SS    66       // slabT column stride (floats), even -> 8B aligned frags
#define KC    64       // K chunk rows per slab
#define BIGF  1e30f

typedef float v2f __attribute__((ext_vector_type(2)));
typedef float v8f __attribute__((ext_vector_type(8)));

__device__ __forceinline__ float wave_red_sum(float v) {
  // wave32 xor-tree; all lanes end with the total
  #pragma unroll
  for (int m = 16; m > 0; m >>= 1) v += __shfl_xor(v, m, 32);
  return v;
}

__global__ __launch_bounds__(256)
void AlphaChebyshevProjection_84000970375198_kernel(
    const float* __restrict__ xh, const float* __restrict__ A,
    const float* __restrict__ b, float* __restrict__ out) {

  __shared__ float slabT[NTI * 16 * SS];   // 144*66 floats = 38016 B
  __shared__ float Hm[130 * HS];           // 130*131 floats = 68120 B
  __shared__ float normv[Mm];
  __shared__ float sv[Mm];
  __shared__ float invsv[Mm];
  __shared__ float zv[NZ + 3];
  __shared__ float dzv[NZ + 3];
  __shared__ float rhs[NZ + 3];
  __shared__ float red[256];

  const int p    = blockIdx.x;
  const float* Ap = A  + (size_t)p * Mm * Nn;
  const float* bp = b  + (size_t)p * Mm;
  const float* xp = xh + (size_t)p * Nn;
  float*       op = out + (size_t)p * Nn;

  const int tid  = threadIdx.x;
  const int lane = tid & 31;
  const int wave = tid >> 5;
  const int half = lane >> 4;
  const int lm   = lane & 15;

  // ---- static tile assignment: pad 45 upper-triangle tiles to 48 = 8 waves x 6
  // all q-indexed arrays are only touched inside fully-unrolled q loops -> SROA
  int  tis[6], tjs[6];
  bool act[6];
  #pragma unroll
  for (int q = 0; q < 6; ++q) {
    int t = wave * 6 + q;
    bool a = (t < 45);
    int tt = a ? t : 0;
    int ti = 0, rem = tt;
    while (rem >= (NTI - ti)) { rem -= (NTI - ti); ++ti; }
    tis[q] = ti; tjs[q] = ti + rem; act[q] = a;
  }

  // ---- phase 0: row norms, row sums, strictly feasible r0 ----
  float lmin = BIGF;
  for (int r = wave * 64; r < wave * 64 + 64; ++r) {
    float s2 = 0.f, s1 = 0.f;
    #pragma unroll
    for (int k = 0; k < 4; ++k) {
      float a = Ap[r * Nn + lane + 32 * k];
      s2 += a * a; s1 += a;
    }
    s2 = wave_red_sum(s2);
    s1 = wave_red_sum(s1);
    float nrm = sqrtf(s2);
    if (lane == 0) normv[r] = nrm;
    float s0 = bp[r] - 0.05f * s1;          // slack at x = X_INIT anchor
    lmin = fminf(lmin, s0 / nrm);
  }
  __syncthreads();
  red[tid] = lmin; __syncthreads();
  for (int off = 128; off; off >>= 1) {
    if (tid < off) red[tid] = fminf(red[tid], red[tid + off]);
    __syncthreads();
  }
  float r0 = 0.5f * red[0];
  for (int j = tid; j < NZ; j += 256) zv[j] = (j < Nn) ? 0.05f : r0;
  __syncthreads();

  // ---- Newton barrier loop: 7 outer (t *= 10) x 10 inner ----
  float t_bar = 1.0f;
  for (int outer = 0; outer < 7; ++outer, t_bar *= 10.0f) {
    for (int inner = 0; inner < 10; ++inner) {

      // slacks of the top (A) rows: s = b - A x - r*||A_i||
      for (int r = wave * 64; r < wave * 64 + 64; ++r) {
        float d0 = 0.f;
        #pragma unroll
        for (int k = 0; k < 4; ++k) { int c = lane + 32 * k; d0 += Ap[r * Nn + c] * zv[c]; }
        d0 = wave_red_sum(d0);
        float s = bp[r] - (d0 + normv[r] * zv[128]);
        if (lane == 0) { sv[r] = s; invsv[r] = 1.0f / s; }
      }

      // ---- Hessian SYRK via WMMA: Hext = (diag(1/s)[A|norm|1|0pad])^T * same ----
      v8f acc[6];
      const v8f vzero = {0.f,0.f,0.f,0.f,0.f,0.f,0.f,0.f};
      #pragma unroll
      for (int q = 0; q < 6; ++q) acc[q] = vzero;

      for (int kb = 0; kb < Mm; kb += KC) {
        __syncthreads();                       // slab WAR fence
        // cooperative slab fill, column-major slabT[col][k]
        for (int idx = tid; idx < KC * 36; idx += 256) {
          int rl = idx / 36;
          int c4 = (idx - rl * 36) * 4;
          int row = kb + rl;
          float v0, v1, v2, v3;
          if (c4 < Nn) {
            const float4 av = *(const float4*)(Ap + (size_t)row * Nn + c4);
            float w = invsv[row];
            v0 = av.x * w; v1 = av.y * w; v2 = av.z * w; v3 = av.w * w;
          } else if (c4 == Nn) {
            v0 = invsv[row] * normv[row];      // col 128: scaled norms
            v1 = 1.0f;                         // col 129: ones -> g = A^T (1/s)
            v2 = 0.f; v3 = 0.f;
          } else { v0 = v1 = v2 = v3 = 0.f; }
          slabT[(c4 + 0) * SS + rl] = v0;
          slabT[(c4 + 1) * SS + rl] = v1;
          slabT[(c4 + 2) * SS + rl] = v2;
          slabT[(c4 + 3) * SS + rl] = v3;
        }
        __syncthreads();
        // per-wave tile MACs over this K chunk (fully unrolled; acc stays in VGPRs)
        #pragma unroll
        for (int q = 0; q < 6; ++q) {
          if (act[q]) {
            const float* pa = slabT + (tis[q] * 16 + lm) * SS;
            const float* pb = slabT + (tjs[q] * 16 + lm) * SS;
            #pragma unroll
            for (int k0 = 0; k0 < KC; k0 += 4) {
              int kk = k0 + 2 * half;          // A 16x4 / B 4x16 fp32 lane layout
              v2f fa = *(const v2f*)(pa + kk);
              v2f fb = *(const v2f*)(pb + kk);
              acc[q] = __builtin_amdgcn_wmma_f32_16x16x4_f32(
                  false, fa, false, fb, (short)0, acc[q], false, false);
            }
          }
        }
      }
      __syncthreads();

      // tile writeback (+ mirror strictly-lower blocks)
      #pragma unroll
      for (int q = 0; q < 6; ++q) {
        if (act[q]) {
          int ti = tis[q], tj = tjs[q];
          #pragma unroll
          for (int v = 0; v < 8; ++v) {
            int rr = ti * 16 + v + 8 * half;
            int cc = tj * 16 + lm;
            float val = acc[q][v];
            if (rr < 130 && cc < 130) Hm[rr * HS + cc] = val;
            if (ti != tj && rr < NZ && cc < NZ) Hm[cc * HS + rr] = val;
          }
        }
      }
      __syncthreads();

      // diagonal (x>=0 barrier + ridge) and gradient rhs = -g
      for (int j = tid; j < NZ; j += 256) {
        float zj = zv[j];
        float invb = (j < Nn) ? 1.0f / zj : 0.0f;
        Hm[j * HS + j] += invb * invb + 1e-8f;
        float g = Hm[j * HS + 129] - invb + ((j == 128) ? -t_bar : 0.0f);
        rhs[j] = -g;
      }
      __syncthreads();

      // ---- single-wave (lockstep, barrier-free) Cholesky + triangular solves ----
      if (wave == 0) {
        for (int k = 0; k < NZ; ++k) {
          float lkk = sqrtf(Hm[k * HS + k]);   // uniform across lanes
          float invl = 1.0f / lkk;
          if (lane == 0) Hm[k * HS + k] = lkk;
          for (int i = k + 1 + lane; i < NZ; i += 32) Hm[i * HS + k] *= invl;
          for (int i = k + 1 + lane; i < NZ; i += 32) {
            float lik = Hm[i * HS + k];
            for (int j = k + 1; j <= i; ++j) Hm[i * HS + j] -= lik * Hm[j * HS + k];
          }
        }
        // forward solve L y = rhs
        for (int k = 0; k < NZ; ++k) {
          float yk = rhs[k] / Hm[k * HS + k];
          if (lane == 0) rhs[k] = yk;
          for (int i = k + 1 + lane; i < NZ; i += 32) rhs[i] -= Hm[i * HS + k] * yk;
        }
        // back solve L^T dz = y  (dz left in rhs)
        for (int k = NZ - 1; k >= 0; --k) {
          float xk = rhs[k] / Hm[k * HS + k];
          if (lane == 0) rhs[k] = xk;
          for (int i = lane; i < k; i += 32) rhs[i] -= Hm[k * HS + i] * xk;
        }
      }
      __syncthreads();

      // ---- fraction-to-boundary step ----
      float rmin = BIGF;
      for (int r = wave * 64; r < wave * 64 + 64; ++r) {
        float d0 = 0.f;
        #pragma unroll
        for (int k = 0; k < 4; ++k) { int c = lane + 32 * k; d0 += Ap[r * Nn + c] * rhs[c]; }
        d0 = wave_red_sum(d0);
        float gdz = d0 + normv[r] * rhs[128];
        float ratio = (gdz > 0.f) ? sv[r] / gdz : BIGF;
        rmin = fminf(rmin, ratio);
      }
      for (int j = tid; j < Nn; j += 256) {
        float dzj = rhs[j];                     // bottom rows: Gdz = -dz, s = x
        if (dzj < 0.f) rmin = fminf(rmin, zv[j] / (-dzj));
      }
      __syncthreads();
      red[tid] = rmin; __syncthreads();
      for (int off = 128; off; off >>= 1) {
        if (tid < off) red[tid] = fminf(red[tid], red[tid + off]);
        __syncthreads();
      }
      float step = fminf(1.0f, 0.99f * red[0]);
      __syncthreads();
      for (int j = tid; j < NZ; j += 256) zv[j] += step * rhs[j];
      __syncthreads();
    }
  }

  // ---- alpha-map along ray x0 -> x_hat ----
  for (int j = tid; j < Nn; j += 256) dzv[j] = xp[j] - zv[j];
  __syncthreads();
  float amin = BIGF;
  for (int r = wave * 64; r < wave * 64 + 64; ++r) {
    float dx0 = 0.f, dd = 0.f;
    #pragma unroll
    for (int k = 0; k < 4; ++k) {
      int c = lane + 32 * k;
      float a = Ap[r * Nn + c];
      dx0 += a * zv[c];
      dd  += a * dzv[c];
    }
    dx0 = wave_red_sum(dx0);
    dd  = wave_red_sum(dd);
    float slack = fmaxf(bp[r] - dx0, 0.0f);
    float ai = (dd > 0.f) ? slack / (dd + 1e-12f) : BIGF;
    amin = fminf(amin, ai);
  }
  __syncthreads();
  red[tid] = amin; __syncthreads();
  for (int off = 128; off; off >>= 1) {
    if (tid < off) red[tid] = fminf(red[tid], red[tid + off]);
    __syncthreads();
  }
  float alpha = fminf(fmaxf(red[0] - 1e-9f, 0.0f), 1.0f);
  for (int j = tid; j < Nn; j += 256) op[j] = zv[j] + alpha * dzv[j];
}

extern "C" void kernel_launch(void* const* d_in, const int* in_sizes, int n_in,
                              void* d_out, int out_size, void* d_ws, size_t ws_size,
                              hipStream_t stream) {
  const float* xh = (const float*)d_in[0];   // [B,S,128]
  const float* A  = (const float*)d_in[1];   // [B,S,512,128]
  const float* b  = (const float*)d_in[2];   // [B,S,512]
  float* out = (float*)d_out;                // [B,S,128]
  int bs = in_sizes[0] / Nn;                 // 512 problems
  AlphaChebyshevProjection_84000970375198_kernel<<<bs, 256, 0, stream>>>(xh, A, b, out);
}